// LogitCompression_52338471469100
// MI455X (gfx1250) — compile-verified
//
#include <hip/hip_runtime.h>
#include <stdint.h>

// LogitCompression on MI455X (gfx1250).
// Latency/transcendental-bound problem: ~1MB HBM traffic (≈45ns @ 23.3TB/s),
// ~102M exp evaluations. No GEMM structure -> WMMA inapplicable; instead use
// the CDNA5 async global->LDS DMA path (ASYNCcnt), wave32 shuffle reductions,
// LDS broadcast reads, and ds_min atomics.

#define ROW_C   1000
#define N_TEMPS 800
#define TPB     800   // 25 wave32 waves; exactly one temperature per thread

__global__ __launch_bounds__(TPB) void
logit_compress_kernel(const float* __restrict__ logits,
                      float* __restrict__ out)
{
    __shared__ __align__(16) float s_d[ROW_C];   // raw logits, then (l - m)
    __shared__ float s_wred[32];                 // per-wave maxes (25 used)
    __shared__ float s_m;
    __shared__ float s_conf;
    __shared__ int   s_minj;

    const int row = blockIdx.x;
    const int tid = threadIdx.x;
    const float* __restrict__ rowp = logits + (size_t)row * ROW_C;

    // ---- stage row into LDS via async DMA (gfx1250 ASYNCcnt path) ----
    // 250 lanes each move 16B; row start is 16B aligned (4000B rows).
    if (tid < ROW_C / 4) {
        // low 32 bits of the generic pointer to a __shared__ object are the
        // wave-relative LDS byte address (ISA 10.2 aperture mapping).
        unsigned           lds = (unsigned)(uint64_t)(uintptr_t)&s_d[tid * 4];
        unsigned long long ga  = (unsigned long long)(uintptr_t)(rowp + tid * 4);
        asm volatile("global_load_async_to_lds_b128 %0, %1, off"
                     :: "v"(lds), "v"(ga) : "memory");
    }
    asm volatile("s_wait_asynccnt 0x0" ::: "memory");
    __syncthreads();

    // ---- row max: wave32 shuffle reduce, then cross-wave via LDS ----
    float lm = -3.402823466e38f;
    for (int i = tid; i < ROW_C; i += TPB) lm = fmaxf(lm, s_d[i]);
    #pragma unroll
    for (int off = 16; off > 0; off >>= 1)
        lm = fmaxf(lm, __shfl_xor(lm, off, 32));
    if ((tid & 31) == 0) s_wred[tid >> 5] = lm;
    __syncthreads();
    if (tid == 0) {
        float m = s_wred[0];
        for (int w = 1; w < TPB / 32; ++w) m = fmaxf(m, s_wred[w]);
        s_m = m;
        s_minj = 0x7fffffff;
    }
    __syncthreads();
    const float m = s_m;
    for (int i = tid; i < ROW_C; i += TPB) s_d[i] = __fsub_rn(s_d[i], m);
    __syncthreads();

    // ---- one temperature per thread: S(t) = sum_i exp((l_i - m)/t) ----
    // temps[j] = 1.0 - 0.00125*j, computed with forced separate roundings
    // to match jnp's mul-then-sub exactly. conf_t = 1/S (t0 == 1.0 gives the
    // original max-softmax confidence for free on thread 0).
    const int   j = tid;
    const float t = __fsub_rn(1.0f, __fmul_rn(0.00125f, (float)j));
    const float r = 1.0f / t;
    const float4* __restrict__ d4 = (const float4*)s_d;
    float a0 = 0.f, a1 = 0.f, a2 = 0.f, a3 = 0.f;
    for (int i = 0; i < ROW_C / 4; ++i) {
        float4 v = d4[i];                 // all lanes same addr -> LDS broadcast
        a0 += expf(v.x * r);
        a1 += expf(v.y * r);
        a2 += expf(v.z * r);
        a3 += expf(v.w * r);
    }
    const float S      = (a0 + a1) + (a2 + a3);
    const float conf_j = 1.0f / S;
    if (j == 0) s_conf = conf_j;
    __syncthreads();

    // ---- 3-bin lookup + target confidence (reference f32 constants) ----
    const float conf      = s_conf;
    const float lowers[3] = {0.0f, 1.0f / 3.0f, 2.0f / 3.0f};
    const float uppers[3] = {1.0f / 3.0f, 2.0f / 3.0f, 1.0f};
    const float newlow[3] = {0.8f, 0.86666666666f, 0.93333333333f};
    int b = 0;                                   // argmax-first-true, default 0
    #pragma unroll
    for (int k = 2; k >= 0; --k)
        if (uppers[k] > conf && lowers[k] <= conf) b = k;
    const float new_conf =
        __fadd_rn(newlow[b],
                  __fmul_rn(1.0f / 15.0f,
                            __fdiv_rn(__fsub_rn(conf, lowers[b]),
                                      0.2666666667f)));

    // ---- first temperature index within tolerance (ds_min atomic) ----
    if (fabsf(__fsub_rn(conf_j, new_conf)) <= 0.01f)
        atomicMin(&s_minj, j);
    __syncthreads();

    const int   jmin  = s_minj;
    const bool  found = (jmin < N_TEMPS);
    const float tc    = found
        ? __fsub_rn(1.0f, __fmul_rn(0.00125f, (float)jmin))
        : 1.0f;                                  // divide by 1.0 is exact
    float* __restrict__ orow = out + (size_t)row * ROW_C;
    for (int i = tid; i < ROW_C; i += TPB)
        orow[i] = __fdiv_rn(rowp[i], tc);
}

extern "C" void kernel_launch(void* const* d_in, const int* in_sizes, int n_in,
                              void* d_out, int out_size, void* d_ws, size_t ws_size,
                              hipStream_t stream)
{
    (void)n_in; (void)out_size; (void)d_ws; (void)ws_size;
    const float* logits = (const float*)d_in[0];
    float*       out    = (float*)d_out;
    const int    B      = in_sizes[0] / ROW_C;   // 128 rows of 1000 classes
    logit_compress_kernel<<<dim3(B), dim3(TPB), 0, stream>>>(logits, out);
}